// ResidueProjection_23278722745103
// MI455X (gfx1250) — compile-verified
//
#include <hip/hip_runtime.h>
#include <stdint.h>

// ---------------------------------------------------------------------------
// ResidueProjection for MI455X (gfx1250, wave32)
//
// Per residue: H = P^T Qc  (Qc centered template => centering of P cancels),
// 3x3 SVD via Jacobi eigensolve of H^T H (V) + Gram-Schmidt of H*V (U, S),
// R = Vh diag(1,1,det(Vh U^T)) U^T  (faithful to reference), project template.
//
// Data movement: Tensor Data Mover (TDM) stages 12KB tiles of positions into
// LDS, double-buffered, one issuing wave per block, s_wait_tensorcnt + barrier.
// H-GEMM: V_WMMA_F32_16X16X4_F32, 5 residues per WMMA (A = stacked P_r^T read
// straight from the TDM tile in WMMA A-layout, B = shared Qc constants).
// ---------------------------------------------------------------------------

#define TILE_RES    256                    // residues per tile == blockDim.x
#define TILE_FLOATS (TILE_RES * 12)        // 3072 floats = 12 KB per tile
#define HS_STRIDE   36                     // residues per wave in H scratch
                                           // (32 + 4 slack for garbage rows)

#if defined(__HIP_DEVICE_COMPILE__) && defined(__gfx1250__) && \
    __has_builtin(__builtin_amdgcn_tensor_load_to_lds)
#define HAVE_TDM 1
#else
#define HAVE_TDM 0
#endif

#if HAVE_TDM && __has_builtin(__builtin_amdgcn_wmma_f32_16x16x4_f32)
#define USE_WMMA 1
#else
#define USE_WMMA 0
#endif

#if __has_include(<hip/amd_detail/amd_gfx1250_TDM.h>)
#define TDM_6ARG 1   // amdgpu-toolchain / therock headers -> 6-arg builtin
#else
#define TDM_6ARG 0   // ROCm 7.2 -> 5-arg builtin
#endif

typedef __attribute__((ext_vector_type(4))) unsigned int u32x4;
typedef __attribute__((ext_vector_type(8))) int          i32x8;
typedef __attribute__((ext_vector_type(4))) int          i32x4;
typedef __attribute__((ext_vector_type(2))) float        v2f;
typedef __attribute__((ext_vector_type(8))) float        v8f;

#if HAVE_TDM
// Issue a 1-D TDM load of `n` floats starting at `gptr` into LDS offset
// `lds_off`. D# layout per CDNA5 ISA ch.8 (group0 128b, group1 256b).
__device__ __forceinline__ void tdm_issue_1d(const float* gptr, uint32_t n,
                                             uint32_t lds_off) {
  uint64_t gaddr = (uint64_t)(uintptr_t)gptr;

  u32x4 g0;
  g0[0] = 1u;                                    // count=1 (valid), user mode
  g0[1] = lds_off;                               // lds_addr (bytes, wave-rel)
  g0[2] = (uint32_t)gaddr;                       // global_addr[31:0]
  g0[3] = (uint32_t)((gaddr >> 32) & 0x01FFFFFFu) | (2u << 30); // ga[56:32]|type=2

  i32x8 g1;
  g1[0] = (int)(2u << 16);                       // data_size=4B; mask/pad/iter=0
  g1[1] = (int)((n & 0xFFFFu) << 16);            // tensor_dim0[15:0]
  g1[2] = (int)((n >> 16) | (1u << 16));         // tensor_dim0[31:16] | tdim1=1
  g1[3] = (int)((n & 0xFFFFu) << 16);            // tile_dim0 = n  (n <= 3072)
  g1[4] = 0;                                     // tile_dim1=0, tile_dim2=0
  g1[5] = (int)n;                                // tensor_dim0_stride[31:0]
  g1[6] = 0;                                     // stride0 hi | stride1 lo
  g1[7] = 0;                                     // stride1 hi

  i32x4 gz = {0, 0, 0, 0};
#if TDM_6ARG
  i32x8 gz8 = {0, 0, 0, 0, 0, 0, 0, 0};
  __builtin_amdgcn_tensor_load_to_lds(g0, g1, gz, gz, gz8, 0);
#else
  __builtin_amdgcn_tensor_load_to_lds(g0, g1, gz, gz, 0);
#endif
}
#endif  // HAVE_TDM

// One classical Jacobi rotation on symmetric A (pivot p,q), accumulating V.
#define JROT(app, aqq, apq, apk, aqk, vp0, vp1, vp2, vq0, vq1, vq2)        \
  do {                                                                     \
    float _q = apq;                                                        \
    if (__builtin_fabsf(_q) > 1e-30f) {                                    \
      float _tau = (aqq - app) * 0.5f / _q;                                \
      float _t = ((_tau >= 0.f) ? 1.f : -1.f) /                            \
                 (__builtin_fabsf(_tau) + sqrtf(1.f + _tau * _tau));       \
      float _c = rsqrtf(1.f + _t * _t);                                    \
      float _s = _t * _c;                                                  \
      app -= _t * _q; aqq += _t * _q; apq = 0.f;                           \
      float _pk = apk, _qk = aqk;                                          \
      apk = _c * _pk - _s * _qk; aqk = _s * _pk + _c * _qk;                \
      float _x;                                                            \
      _x = vp0; vp0 = _c * _x - _s * vq0; vq0 = _s * _x + _c * vq0;        \
      _x = vp1; vp1 = _c * _x - _s * vq1; vq1 = _s * _x + _c * vq1;        \
      _x = vp2; vp2 = _c * _x - _s * vq2; vq2 = _s * _x + _c * vq2;        \
    }                                                                      \
  } while (0)

// Conditional descending swap of (eigenvalue, V column) pairs.
#define CSWAP(la, lb, x0, x1, x2, y0, y1, y2)                              \
  do {                                                                     \
    if (la < lb) {                                                         \
      float _t;                                                            \
      _t = la; la = lb; lb = _t;                                           \
      _t = x0; x0 = y0; y0 = _t;                                           \
      _t = x1; x1 = y1; y1 = _t;                                           \
      _t = x2; x2 = y2; y2 = _t;                                           \
    }                                                                      \
  } while (0)

__global__ void __launch_bounds__(TILE_RES)
ResidueProjection_23278722745103_kernel(const float* __restrict__ positions,
                                        const float* __restrict__ tmpl,
                                        float* __restrict__ out,
                                        int numResidues, int numTiles,
                                        int tilesPerBlock) {
#if HAVE_TDM
  __shared__ __align__(16) float smem[2 * TILE_FLOATS];     // 24 KB dbl buffer
#endif
#if USE_WMMA
  __shared__ __align__(16) float smemH[8 * HS_STRIDE * 12]; // 13.5 KB H scratch
#endif
  const int tid  = threadIdx.x;
  const int lane = tid & 31;
  const int wv   = tid >> 5;
  (void)lane; (void)wv;

  // --- template (uniform, scalar-cached loads) ---
  float T[12];
#pragma unroll
  for (int k = 0; k < 12; ++k) T[k] = tmpl[k];
  const float qm0 = 0.25f * (T[0] + T[3] + T[6] + T[9]);
  const float qm1 = 0.25f * (T[1] + T[4] + T[7] + T[10]);
  const float qm2 = 0.25f * (T[2] + T[5] + T[8] + T[11]);
  float Qc[4][3];
#pragma unroll
  for (int a = 0; a < 4; ++a) {
    Qc[a][0] = T[a * 3 + 0] - qm0;
    Qc[a][1] = T[a * 3 + 1] - qm1;
    Qc[a][2] = T[a * 3 + 2] - qm2;
  }

#if USE_WMMA
  // --- per-lane WMMA layout constants (computed once) ---
  const int nCol  = lane & 15;        // N column / A row (m)
  const int half  = lane >> 4;        // 0: K=0,1 ; 1: K=2,3
  const int rSlot = nCol / 3;         // residue slot of A row m
  const int iSlot = nCol - 3 * rSlot; // i-index of A row m
  const int aOff0 = (half * 2 + 0) * 3 + iSlot;   // element offset a*3+i
  const int aOff1 = (half * 2 + 1) * 3 + iSlot;

  // B[k][n] = Qc[k][n] for n<3 else 0 ; lane holds rows k = 2*half{+0,+1}
  float bv0 = 0.f, bv1 = 0.f;
  if (nCol < 3) {
    float q0n = (nCol == 0) ? Qc[0][0] : (nCol == 1) ? Qc[0][1] : Qc[0][2];
    float q1n = (nCol == 0) ? Qc[1][0] : (nCol == 1) ? Qc[1][1] : Qc[1][2];
    float q2n = (nCol == 0) ? Qc[2][0] : (nCol == 1) ? Qc[2][1] : Qc[2][2];
    float q3n = (nCol == 0) ? Qc[3][0] : (nCol == 1) ? Qc[3][1] : Qc[3][2];
    bv0 = half ? q2n : q0n;
    bv1 = half ? q3n : q1n;
  }

  // D scatter constants: VGPR v holds D row m = v + 8*half = H[rv][iv] (col n)
  int sOff[8];
#pragma unroll
  for (int v = 0; v < 8; ++v) {
    const int rv = half ? ((v + 8) / 3) : (v / 3);
    const int iv = half ? ((v + 8) % 3) : (v % 3);
    sOff[v] = rv * 12 + iv * 3;       // + nCol added via base pointer
  }
#endif  // USE_WMMA

  const int tileBeg = blockIdx.x * tilesPerBlock;
  int tileEnd = tileBeg + tilesPerBlock;
  if (tileEnd > numTiles) tileEnd = numTiles;
  if (tileBeg >= tileEnd) return;

#if HAVE_TDM
  // Prologue: issuing wave (wave 0) kicks off tile 0's DMA into buffer 0.
  if (tid < 32) {
    long f0 = (long)tileBeg * TILE_FLOATS;
    long rem = (long)numResidues * 12 - f0;
    uint32_t n = (uint32_t)(rem < TILE_FLOATS ? rem : TILE_FLOATS);
    tdm_issue_1d(positions + f0, n, (uint32_t)(uintptr_t)&smem[0]);
  }
#endif

  for (int t = tileBeg; t < tileEnd; ++t) {
    const int bufSel = (t - tileBeg) & 1;

#if HAVE_TDM
    if (tid < 32) {
#if __has_builtin(__builtin_amdgcn_s_wait_tensorcnt)
      __builtin_amdgcn_s_wait_tensorcnt(0);      // tile t landed in LDS
#else
      asm volatile("s_wait_tensorcnt 0x0" ::: "memory");
#endif
    }
    __syncthreads();                             // hand tile t to all 8 waves
    if (tid < 32 && (t + 1) < tileEnd) {         // prefetch tile t+1 via TDM
      long f0 = (long)(t + 1) * TILE_FLOATS;
      long rem = (long)numResidues * 12 - f0;
      uint32_t n = (uint32_t)(rem < TILE_FLOATS ? rem : TILE_FLOATS);
      tdm_issue_1d(positions + f0, n,
                   (uint32_t)(uintptr_t)&smem[(bufSel ^ 1) * TILE_FLOATS]);
    }
#endif

#if USE_WMMA
    // ---- H-GEMM on the matrix pipe: 7 WMMAs cover this wave's 32 residues,
    // 5 residues per V_WMMA_F32_16X16X4_F32 (shared B = Qc). EXEC is all-1s
    // here (no divergent guard encloses this phase).
    {
      const float* tile = &smem[bufSel * TILE_FLOATS + wv * 32 * 12];
      const v2f B = {bv0, bv1};
#pragma unroll
      for (int g = 0; g < 7; ++g) {
        int ra = 5 * g + rSlot;                  // A-row residue (clamped)
        if (ra > 31) ra = 31;                    // garbage rows read residue 31
        const float* pa = tile + ra * 12;
        const v2f A = {pa[aOff0], pa[aOff1]};
        v8f D = {};
        D = __builtin_amdgcn_wmma_f32_16x16x4_f32(
            /*neg_a=*/false, A, /*neg_b=*/false, B,
            /*c_mod=*/(short)0, D, /*reuse_a=*/false, /*reuse_b=*/false);
        // Scatter H entries; slots past residue 31 land in per-wave slack,
        // m=15 garbage is overwritten in-order by the next group.
        float* hs = &smemH[(wv * HS_STRIDE + 5 * g) * 12 + nCol];
        if (nCol < 3) {
#pragma unroll
          for (int v = 0; v < 8; ++v) hs[sOff[v]] = D[v];
        }
      }
    }
#endif  // USE_WMMA

    const int r = t * TILE_RES + tid;            // global residue index
    if (r < numResidues) {
      float4 P0, P1, P2;
#if HAVE_TDM
      const float4* src = (const float4*)&smem[bufSel * TILE_FLOATS];
      P0 = src[tid * 3 + 0];
      P1 = src[tid * 3 + 1];
      P2 = src[tid * 3 + 2];
#else
      const float4* src = (const float4*)(positions + (size_t)r * 12);
      P0 = src[0]; P1 = src[1]; P2 = src[2];
#endif
      const float Pm[4][3] = {{P0.x, P0.y, P0.z}, {P0.w, P1.x, P1.y},
                              {P1.z, P1.w, P2.x}, {P2.y, P2.z, P2.w}};

      const bool isZero =
          (P0.x == 0.f) & (P0.y == 0.f) & (P0.z == 0.f) & (P0.w == 0.f) &
          (P1.x == 0.f) & (P1.y == 0.f) & (P1.z == 0.f) & (P1.w == 0.f) &
          (P2.x == 0.f) & (P2.y == 0.f) & (P2.z == 0.f) & (P2.w == 0.f);

      const float pm0 = 0.25f * (Pm[0][0] + Pm[1][0] + Pm[2][0] + Pm[3][0]);
      const float pm1 = 0.25f * (Pm[0][1] + Pm[1][1] + Pm[2][1] + Pm[3][1]);
      const float pm2 = 0.25f * (Pm[0][2] + Pm[1][2] + Pm[2][2] + Pm[3][2]);

      // H = Pc^T Qc == P^T Qc (Qc columns sum to zero)
      float H[3][3];
#if USE_WMMA
      {
        const float* hv = &smemH[(wv * HS_STRIDE + lane) * 12];
        const float4 h0 = ((const float4*)hv)[0];
        const float4 h1 = ((const float4*)hv)[1];
        const float  h8 = hv[8];
        H[0][0] = h0.x; H[0][1] = h0.y; H[0][2] = h0.z;
        H[1][0] = h0.w; H[1][1] = h1.x; H[1][2] = h1.y;
        H[2][0] = h1.z; H[2][1] = h1.w; H[2][2] = h8;
      }
#else
#pragma unroll
      for (int i = 0; i < 3; ++i)
#pragma unroll
        for (int j = 0; j < 3; ++j)
          H[i][j] = Pm[0][i] * Qc[0][j] + Pm[1][i] * Qc[1][j] +
                    Pm[2][i] * Qc[2][j] + Pm[3][i] * Qc[3][j];
#endif

      // A = H^T H (symmetric) -> Jacobi eigensolve gives V, lambda = S^2
      float a00 = H[0][0]*H[0][0] + H[1][0]*H[1][0] + H[2][0]*H[2][0];
      float a11 = H[0][1]*H[0][1] + H[1][1]*H[1][1] + H[2][1]*H[2][1];
      float a22 = H[0][2]*H[0][2] + H[1][2]*H[1][2] + H[2][2]*H[2][2];
      float a01 = H[0][0]*H[0][1] + H[1][0]*H[1][1] + H[2][0]*H[2][1];
      float a02 = H[0][0]*H[0][2] + H[1][0]*H[1][2] + H[2][0]*H[2][2];
      float a12 = H[0][1]*H[0][2] + H[1][1]*H[1][2] + H[2][1]*H[2][2];

      float v00 = 1.f, v01 = 0.f, v02 = 0.f;
      float v10 = 0.f, v11 = 1.f, v12 = 0.f;
      float v20 = 0.f, v21 = 0.f, v22 = 1.f;

#pragma unroll
      for (int sweep = 0; sweep < 4; ++sweep) {
        JROT(a00, a11, a01, a02, a12, v00, v10, v20, v01, v11, v21);
        JROT(a00, a22, a02, a01, a12, v00, v10, v20, v02, v12, v22);
        JROT(a11, a22, a12, a01, a02, v01, v11, v21, v02, v12, v22);
      }

      // sort eigenpairs descending (matches LAPACK singular-value order)
      CSWAP(a00, a11, v00, v10, v20, v01, v11, v21);
      CSWAP(a00, a22, v00, v10, v20, v02, v12, v22);
      CSWAP(a11, a22, v01, v11, v21, v02, v12, v22);

      // B = H*V ; columns b_j = s_j * u_j
      float b0x = H[0][0]*v00 + H[0][1]*v10 + H[0][2]*v20;
      float b0y = H[1][0]*v00 + H[1][1]*v10 + H[1][2]*v20;
      float b0z = H[2][0]*v00 + H[2][1]*v10 + H[2][2]*v20;
      float b1x = H[0][0]*v01 + H[0][1]*v11 + H[0][2]*v21;
      float b1y = H[1][0]*v01 + H[1][1]*v11 + H[1][2]*v21;
      float b1z = H[2][0]*v01 + H[2][1]*v11 + H[2][2]*v21;
      float b2x = H[0][0]*v02 + H[0][1]*v12 + H[0][2]*v22;
      float b2y = H[1][0]*v02 + H[1][1]*v12 + H[1][2]*v22;
      float b2z = H[2][0]*v02 + H[2][1]*v12 + H[2][2]*v22;

      // U via Gram-Schmidt with rank-deficiency fallbacks
      float n0 = b0x*b0x + b0y*b0y + b0z*b0z;
      float inv0 = rsqrtf(fmaxf(n0, 1e-30f));
      float u0x = b0x*inv0, u0y = b0y*inv0, u0z = b0z*inv0;

      float d01 = u0x*b1x + u0y*b1y + u0z*b1z;
      float w1x = b1x - d01*u0x, w1y = b1y - d01*u0y, w1z = b1z - d01*u0z;
      float n1 = w1x*w1x + w1y*w1y + w1z*w1z;
      float u1x, u1y, u1z;
      if (n1 > 1e-20f) {
        float inv1 = rsqrtf(n1);
        u1x = w1x*inv1; u1y = w1y*inv1; u1z = w1z*inv1;
      } else {  // any unit vector orthogonal to u0
        float ex = (__builtin_fabsf(u0x) < 0.9f) ? 1.f : 0.f;
        float ey = 1.f - ex;
        float cx = ey*u0z, cy = -ex*u0z, cz = ex*u0y - ey*u0x;
        float invc = rsqrtf(fmaxf(cx*cx + cy*cy + cz*cz, 1e-30f));
        u1x = cx*invc; u1y = cy*invc; u1z = cz*invc;
      }

      float d02 = u0x*b2x + u0y*b2y + u0z*b2z;
      float d12 = u1x*b2x + u1y*b2y + u1z*b2z;
      float w2x = b2x - d02*u0x - d12*u1x;
      float w2y = b2y - d02*u0y - d12*u1y;
      float w2z = b2z - d02*u0z - d12*u1z;
      float n2 = w2x*w2x + w2y*w2y + w2z*w2z;
      float u2x, u2y, u2z;
      if (n2 > 1e-20f) {
        float inv2 = rsqrtf(n2);
        u2x = w2x*inv2; u2y = w2y*inv2; u2z = w2z*inv2;
      } else {  // s3 ~ 0: complete the basis
        u2x = u0y*u1z - u0z*u1y;
        u2y = u0z*u1x - u0x*u1z;
        u2z = u0x*u1y - u0y*u1x;
      }

      // d = det(Vh U^T) = det(V) * det(U), clamped to +/-1
      float detV = v00*(v11*v22 - v12*v21) - v01*(v10*v22 - v12*v20) +
                   v02*(v10*v21 - v11*v20);
      float detU = u0x*(u1y*u2z - u1z*u2y) - u0y*(u1x*u2z - u1z*u2x) +
                   u0z*(u1x*u2y - u1y*u2x);
      float d = (detV * detU < 0.f) ? -1.f : 1.f;

      // R = Vh diag(1,1,d) U^T : R[i][k] = V[0][i]u0[k]+V[1][i]u1[k]+d V[2][i]u2[k]
      float du2x = d*u2x, du2y = d*u2y, du2z = d*u2z;
      float R[3][3];
      R[0][0] = v00*u0x + v10*u1x + v20*du2x;
      R[0][1] = v00*u0y + v10*u1y + v20*du2y;
      R[0][2] = v00*u0z + v10*u1z + v20*du2z;
      R[1][0] = v01*u0x + v11*u1x + v21*du2x;
      R[1][1] = v01*u0y + v11*u1y + v21*du2y;
      R[1][2] = v01*u0z + v11*u1z + v21*du2z;
      R[2][0] = v02*u0x + v12*u1x + v22*du2x;
      R[2][1] = v02*u0y + v12*u1y + v22*du2y;
      R[2][2] = v02*u0z + v12*u1z + v22*du2z;

      const float t0 = qm0 - (R[0][0]*pm0 + R[0][1]*pm1 + R[0][2]*pm2);
      const float t1 = qm1 - (R[1][0]*pm0 + R[1][1]*pm1 + R[1][2]*pm2);
      const float t2 = qm2 - (R[2][0]*pm0 + R[2][1]*pm1 + R[2][2]*pm2);

      // proj[a][i] = sum_j T[a][j] R[i][j] + t[i]
      float pr[4][3];
#pragma unroll
      for (int a = 0; a < 4; ++a) {
#pragma unroll
        for (int i = 0; i < 3; ++i) {
          pr[a][i] = T[a*3 + 0]*R[i][0] + T[a*3 + 1]*R[i][1] +
                     T[a*3 + 2]*R[i][2] + ((i == 0) ? t0 : (i == 1) ? t1 : t2);
        }
      }

      float4 O0 = make_float4(pr[0][0], pr[0][1], pr[0][2], pr[1][0]);
      float4 O1 = make_float4(pr[1][1], pr[1][2], pr[2][0], pr[2][1]);
      float4 O2 = make_float4(pr[2][2], pr[3][0], pr[3][1], pr[3][2]);
      if (isZero) { O0 = P0; O1 = P1; O2 = P2; }

      float4* dst = (float4*)(out + (size_t)r * 12);
      dst[0] = O0; dst[1] = O1; dst[2] = O2;
    }
  }
}

extern "C" void kernel_launch(void* const* d_in, const int* in_sizes, int n_in,
                              void* d_out, int out_size, void* d_ws,
                              size_t ws_size, hipStream_t stream) {
  (void)n_in; (void)out_size; (void)d_ws; (void)ws_size;
  const float* positions = (const float*)d_in[0];
  // d_in[1] = aa_types (unused by the reference forward)
  const float* tmpl = (const float*)d_in[2];
  float* out = (float*)d_out;

  const int numResidues = in_sizes[0] / 12;
  const int numTiles = (numResidues + TILE_RES - 1) / TILE_RES;
  int grid = numTiles < 2048 ? numTiles : 2048;
  const int tilesPerBlock = (numTiles + grid - 1) / grid;
  grid = (numTiles + tilesPerBlock - 1) / tilesPerBlock;

  hipLaunchKernelGGL(ResidueProjection_23278722745103_kernel, dim3(grid),
                     dim3(TILE_RES), 0, stream, positions, tmpl, out,
                     numResidues, numTiles, tilesPerBlock);
}